// Memory_fingerprint_57217554317328
// MI455X (gfx1250) — compile-verified
//
#include <hip/hip_runtime.h>
#include <math.h>

// ---------------------------------------------------------------------------
// Problem constants (fixed by the reference setup_inputs()).
// ---------------------------------------------------------------------------
#define DIM   512      // feature dim
#define BLK   64       // fingerprint block size (rows per memory context)
#define MCTX  6304     // number of memory contexts
#define BATCH 128      // batch size
#define KSEL  50       // top-k
#define EPS   1e-8f

typedef float v2f __attribute__((ext_vector_type(2)));
typedef float v8f __attribute__((ext_vector_type(8)));

// ---------------------------------------------------------------------------
// Kernel 1: inverse row norms  inv[row] = 1 / max(||x_row||, EPS)
// ---------------------------------------------------------------------------
__global__ void rownorm_inv_kernel(const float* __restrict__ x,
                                   float* __restrict__ inv, int ncols) {
    const int row = blockIdx.x;
    const float* p = x + (size_t)row * ncols;
    float s = 0.f;
    for (int c = threadIdx.x; c < ncols; c += blockDim.x) {
        float v = p[c];
        s += v * v;
    }
    __shared__ float red[256];
    red[threadIdx.x] = s;
    __syncthreads();
    for (int off = blockDim.x >> 1; off > 0; off >>= 1) {
        if ((int)threadIdx.x < off) red[threadIdx.x] += red[threadIdx.x + off];
        __syncthreads();
    }
    if (threadIdx.x == 0) {
        inv[row] = 1.0f / fmaxf(sqrtf(red[0]), EPS);
    }
}

// ---------------------------------------------------------------------------
// Kernel 2: cos_sim[b, m] = <q_b, mc_m> * invq[b] * invm[m]
// One wave (32 lanes) per 16x16 output tile, K-loop of V_WMMA_F32_16X16X4_F32.
// A layout (16x4 f32): lanes 0-15 -> M=lane, K={0,1}; lanes 16-31 -> K={2,3}.
// B layout (4x16 f32): mirrored; N = lane & 15.
// C/D layout: VGPR r -> row r + 8*(lane>=16), col = lane & 15.
// ---------------------------------------------------------------------------
__global__ void cos_gemm_wmma_kernel(const float* __restrict__ q,
                                     const float* __restrict__ mc,
                                     const float* __restrict__ invq,
                                     const float* __restrict__ invm,
                                     float* __restrict__ cosv) {
    const int mTile = blockIdx.x;          // 0..MCTX/16-1
    const int bTile = blockIdx.y;          // 0..BATCH/16-1
    const int lane  = threadIdx.x;         // 0..31 (wave32)
    const int half  = lane >> 4;           // 0 or 1
    const int l15   = lane & 15;
    const int koff  = half * 2;            // K sub-offset for this half-wave

    const float* qp = q  + (size_t)(bTile * 16 + l15) * DIM;   // A row
    const float* mp = mc + (size_t)(mTile * 16 + l15) * DIM;   // B row (= N col)

    v8f acc = {};
    #pragma unroll 8
    for (int kk = 0; kk < DIM; kk += 4) {
        v2f a = *(const v2f*)(qp + kk + koff);
        v2f b = *(const v2f*)(mp + kk + koff);
        // 8 args: (neg_a, A, neg_b, B, c_mod, C, reuse_a, reuse_b)
        acc = __builtin_amdgcn_wmma_f32_16x16x4_f32(
            false, a, false, b, (short)0, acc, false, false);
    }

    const int col = mTile * 16 + l15;
    const float im = invm[col];
    #pragma unroll
    for (int r = 0; r < 8; ++r) {
        const int brow = bTile * 16 + half * 8 + r;
        cosv[(size_t)brow * MCTX + col] = acc[r] * invq[brow] * im;
    }
}

// ---------------------------------------------------------------------------
// Kernel 3: per-batch top-50 (value+index) and softmax over selected values.
// One workgroup per batch row; row cached in LDS (25 KB << 320 KB WGP LDS).
// Deterministic tie-break: smallest index wins.
// ---------------------------------------------------------------------------
__global__ void topk_softmax_kernel(const float* __restrict__ cosv,
                                    float* __restrict__ wout,
                                    int* __restrict__ iout) {
    __shared__ float vals[MCTX];
    __shared__ float rv[256];
    __shared__ int   ri[256];
    __shared__ float tkv[KSEL];

    const int b = blockIdx.x;
    const float* row = cosv + (size_t)b * MCTX;
    for (int m = threadIdx.x; m < MCTX; m += blockDim.x) vals[m] = row[m];
    __syncthreads();

    for (int sel = 0; sel < KSEL; ++sel) {
        float bestv = -INFINITY;
        int   besti = 0x7fffffff;
        for (int m = threadIdx.x; m < MCTX; m += blockDim.x) {
            float v = vals[m];
            if (v > bestv || (v == bestv && m < besti)) { bestv = v; besti = m; }
        }
        rv[threadIdx.x] = bestv;
        ri[threadIdx.x] = besti;
        __syncthreads();
        for (int off = 128; off > 0; off >>= 1) {
            if ((int)threadIdx.x < off) {
                float ov = rv[threadIdx.x + off];
                int   oi = ri[threadIdx.x + off];
                if (ov > rv[threadIdx.x] ||
                    (ov == rv[threadIdx.x] && oi < ri[threadIdx.x])) {
                    rv[threadIdx.x] = ov;
                    ri[threadIdx.x] = oi;
                }
            }
            __syncthreads();
        }
        if (threadIdx.x == 0) {
            tkv[sel] = rv[0];
            iout[b * KSEL + sel] = ri[0];
            vals[ri[0]] = -INFINITY;       // remove winner for next round
        }
        __syncthreads();
    }

    if (threadIdx.x == 0) {                 // 50-element softmax: trivial
        float mx = tkv[0];
        for (int i = 1; i < KSEL; ++i) mx = fmaxf(mx, tkv[i]);
        float s = 0.f;
        for (int i = 0; i < KSEL; ++i) { float e = expf(tkv[i] - mx); tkv[i] = e; s += e; }
        const float is = 1.0f / s;
        for (int i = 0; i < KSEL; ++i) wout[b * KSEL + i] = tkv[i] * is;
    }
}

// ---------------------------------------------------------------------------
// Kernel 4: out[b,l,:] = (sum_k w[b,k] * FP[idx[b,k]*64 + l, :]) * wgt
//                        + enc[b,l,:] * (1 - wgt)
// The bandwidth kernel: ~840 MB of coalesced float4 gather reads.
// grid (BLK, BATCH), 128 threads * float4 = 512 floats of D per block.
// ---------------------------------------------------------------------------
__global__ void gather_blend_kernel(const float* __restrict__ fp,
                                    const float* __restrict__ enc,
                                    const float* __restrict__ w,
                                    const int* __restrict__ idx,
                                    const float* __restrict__ weight,
                                    float* __restrict__ out) {
    const int l = blockIdx.x;   // row within 64-row block
    const int b = blockIdx.y;   // batch

    __shared__ float sw[KSEL];
    __shared__ int   si[KSEL];
    if (threadIdx.x < KSEL) {
        sw[threadIdx.x] = w[b * KSEL + threadIdx.x];
        si[threadIdx.x] = idx[b * KSEL + threadIdx.x];
    }
    __syncthreads();

    const int d4 = threadIdx.x * 4;
    float4 acc = make_float4(0.f, 0.f, 0.f, 0.f);
    #pragma unroll 2
    for (int k = 0; k < KSEL; ++k) {
        const float* p = fp + ((size_t)si[k] * BLK + l) * DIM + d4;
        float4 v = *(const float4*)p;       // global_load_b128, coalesced in D
        const float s = sw[k];
        acc.x += s * v.x; acc.y += s * v.y;
        acc.z += s * v.z; acc.w += s * v.w;
    }

    const float wg  = weight[0];
    const float iwg = 1.0f - wg;
    const size_t o = ((size_t)b * BLK + l) * DIM + d4;
    float4 e = *(const float4*)(enc + o);
    float4 r;
    r.x = acc.x * wg + e.x * iwg;
    r.y = acc.y * wg + e.y * iwg;
    r.z = acc.z * wg + e.z * iwg;
    r.w = acc.w * wg + e.w * iwg;
    *(float4*)(out + o) = r;
}

// ---------------------------------------------------------------------------
// Host launcher
// ---------------------------------------------------------------------------
extern "C" void kernel_launch(void* const* d_in, const int* in_sizes, int n_in,
                              void* d_out, int out_size, void* d_ws, size_t ws_size,
                              hipStream_t stream) {
    const float* enc    = (const float*)d_in[0];   // [B, 64, 512]
    const float* query  = (const float*)d_in[1];   // [B, 512]
    const float* memctx = (const float*)d_in[2];   // [M, 512]
    const float* fp     = (const float*)d_in[3];   // [M*64, 512]
    const float* weight = (const float*)d_in[4];   // [1]
    (void)in_sizes; (void)n_in; (void)out_size; (void)ws_size;

    // Workspace layout (256-byte aligned regions), ~3.3 MB total.
    char* ws = (char*)d_ws;
    size_t off = 0;
    auto alloc = [&](size_t bytes) {
        char* p = ws + off;
        off += (bytes + 255) & ~(size_t)255;
        return p;
    };
    float* cosv = (float*)alloc((size_t)BATCH * MCTX * sizeof(float));
    float* invq = (float*)alloc((size_t)BATCH * sizeof(float));
    float* invm = (float*)alloc((size_t)MCTX * sizeof(float));
    float* tkw  = (float*)alloc((size_t)BATCH * KSEL * sizeof(float));
    int*   tki  = (int*)  alloc((size_t)BATCH * KSEL * sizeof(int));

    // 1) inverse norms
    rownorm_inv_kernel<<<BATCH, 256, 0, stream>>>(query, invq, DIM);
    rownorm_inv_kernel<<<MCTX, 256, 0, stream>>>(memctx, invm, DIM);

    // 2) cosine-similarity GEMM via V_WMMA_F32_16X16X4_F32 (one wave per tile)
    cos_gemm_wmma_kernel<<<dim3(MCTX / 16, BATCH / 16), 32, 0, stream>>>(
        query, memctx, invq, invm, cosv);

    // 3) top-50 + softmax per batch row
    topk_softmax_kernel<<<BATCH, 256, 0, stream>>>(cosv, tkw, tki);

    // 4) weighted block gather + blend (bandwidth-bound, ~840 MB reads)
    gather_blend_kernel<<<dim3(BLK, BATCH), 128, 0, stream>>>(
        fp, enc, tkw, tki, weight, (float*)d_out);
}